// VolumeRenderer_1571958030563
// MI455X (gfx1250) — compile-verified
//
#include <hip/hip_runtime.h>
#include <stdint.h>

#define WAVES_PER_BLOCK 8
#define BLOCK_THREADS (WAVES_PER_BLOCK * 32)

// ---------------------------------------------------------------------------
// CDNA5 async global->LDS path (ASYNCcnt). Guarded: falls back to plain loads
// + global_prefetch_b8 if the builtins aren't declared by this toolchain.
// Probe feedback: builtin params are (global int*, local int*, imm, imm).
// ---------------------------------------------------------------------------
#if defined(__gfx1250__) && \
    __has_builtin(__builtin_amdgcn_global_load_async_to_lds_b32) && \
    __has_builtin(__builtin_amdgcn_s_wait_asynccnt)
#define USE_ASYNC_LDS 1
#else
#define USE_ASYNC_LDS 0
#endif

#if USE_ASYNC_LDS
typedef __attribute__((address_space(1))) int as1_int_t;
typedef __attribute__((address_space(3))) int as3_int_t;
__device__ __forceinline__ as1_int_t* to_as1(const void* p) {
  return (as1_int_t*)p;
}
__device__ __forceinline__ as3_int_t* to_as3(void* p) {
  return (as3_int_t*)p;
}
#endif

__global__ void zero_head_kernel(float* __restrict__ out_total) {
  out_total[0] = 0.0f;
}

// Shared per-chunk math: wave32 inclusive scan of sd, transmittance, weights,
// accumulator updates. `carry` is the running sum of sd over previous chunks.
__device__ __forceinline__ void process_chunk(
    int lane, bool valid, long long g, float sig, float del, float t,
    float r0, float g0, float b0, float thr,
    float& carry, float& acc_op, float& acc_dep,
    float& acc_r, float& acc_g, float& acc_b, int& acc_cnt,
    float* __restrict__ out_w)
{
  float sd = valid ? sig * del : 0.0f;

  // Wave32 inclusive prefix sum (5 shuffle steps).
  float inc = sd;
#pragma unroll
  for (int off = 1; off < 32; off <<= 1) {
    float y = __shfl_up(inc, off, 32);
    if (lane >= off) inc += y;
  }
  float excl = inc - sd;

  float T = __expf(-(carry + excl));      // exclusive transmittance
  float a = 1.0f - __expf(-sd);           // per-sample alpha
  bool alive = T > thr;
  float w = (alive && valid) ? a * T : 0.0f;

  if (valid) out_w[g] = w;

  acc_op  += w;
  acc_dep += w * t;
  acc_r   += w * r0;
  acc_g   += w * g0;
  acc_b   += w * b0;
  acc_cnt += (alive && valid) ? 1 : 0;

  // Broadcast chunk total from lane 31.
  carry += __shfl(inc, 31, 32);
}

__global__ __launch_bounds__(BLOCK_THREADS)
void volrend_fwd_kernel(const float* __restrict__ sigmas,
                        const float* __restrict__ rgbs,
                        const float* __restrict__ deltas,
                        const float* __restrict__ ts,
                        const long long* __restrict__ rays_a,
                        const float* __restrict__ thr_ptr,
                        float* __restrict__ out,  // [total, opacity(B), depth(B), rgb(3B), w(N)]
                        int n_rays)
{
  const int wv   = (int)(threadIdx.x >> 5);
  const int lane = (int)(threadIdx.x & 31);
  const int ray  = (int)blockIdx.x * WAVES_PER_BLOCK + wv;
  if (ray >= n_rays) return;   // whole wave exits together (ray is per-wave)

  const long long ray_id = rays_a[3LL * ray + 0];
  const long long start  = rays_a[3LL * ray + 1];
  const int       count  = (int)rays_a[3LL * ray + 2];
  const float     thr    = *thr_ptr;

  float* __restrict__ out_op  = out + 1;
  float* __restrict__ out_dep = out + 1 + (long long)n_rays;
  float* __restrict__ out_rgb = out + 1 + 2LL * n_rays;
  float* __restrict__ out_w   = out + 1 + 5LL * n_rays;

  float carry = 0.0f;
  float acc_op = 0.0f, acc_dep = 0.0f, acc_r = 0.0f, acc_g = 0.0f, acc_b = 0.0f;
  int   acc_cnt = 0;

#if USE_ASYNC_LDS
  // Per-wave double-buffered staging of the four input streams.
  __shared__ float s_sig[WAVES_PER_BLOCK][2][32];
  __shared__ float s_del[WAVES_PER_BLOCK][2][32];
  __shared__ float s_t  [WAVES_PER_BLOCK][2][32];
  __shared__ float s_rgb[WAVES_PER_BLOCK][2][32 * 3];

  auto stage = [&](int buf, int base) {
    int idx = base + lane;
    long long gc = start + (long long)((idx < count) ? idx : (count - 1));
    __builtin_amdgcn_global_load_async_to_lds_b32(
        to_as1(sigmas + gc), to_as3(&s_sig[wv][buf][lane]), 0, 0);
    __builtin_amdgcn_global_load_async_to_lds_b32(
        to_as1(deltas + gc), to_as3(&s_del[wv][buf][lane]), 0, 0);
    __builtin_amdgcn_global_load_async_to_lds_b32(
        to_as1(ts + gc),     to_as3(&s_t[wv][buf][lane]),   0, 0);
    const float* rg = rgbs + 3LL * gc;
    float* ld = &s_rgb[wv][buf][3 * lane];
    __builtin_amdgcn_global_load_async_to_lds_b32(to_as1(rg + 0), to_as3(ld + 0), 0, 0);
    __builtin_amdgcn_global_load_async_to_lds_b32(to_as1(rg + 1), to_as3(ld + 1), 0, 0);
    __builtin_amdgcn_global_load_async_to_lds_b32(to_as1(rg + 2), to_as3(ld + 2), 0, 0);
  };

  stage(0, 0);
  int buf = 0;
  for (int base = 0; base < count; base += 32) {
    if (base + 32 < count) {
      stage(buf ^ 1, base + 32);                 // 6 more in flight
      __builtin_amdgcn_s_wait_asynccnt(6);       // current chunk's 6 are done
    } else {
      __builtin_amdgcn_s_wait_asynccnt(0);
    }
    asm volatile("" ::: "memory");               // keep LDS reads behind the wait

    int idx = base + lane;
    bool valid = idx < count;
    long long g = start + (long long)(valid ? idx : (count - 1));
    float sig = s_sig[wv][buf][lane];
    float del = s_del[wv][buf][lane];
    float t   = s_t[wv][buf][lane];
    float r0  = s_rgb[wv][buf][3 * lane + 0];
    float g0  = s_rgb[wv][buf][3 * lane + 1];
    float b0  = s_rgb[wv][buf][3 * lane + 2];

    process_chunk(lane, valid, g, sig, del, t, r0, g0, b0, thr,
                  carry, acc_op, acc_dep, acc_r, acc_g, acc_b, acc_cnt, out_w);
    buf ^= 1;
  }
#else
  for (int base = 0; base < count; base += 32) {
    int idx = base + lane;
    bool valid = idx < count;
    long long g = start + (long long)(valid ? idx : (count - 1));

    if (base + 32 < count) {
      // gfx1250 global_prefetch_b8 for the next chunk of each stream.
      __builtin_prefetch(sigmas + g + 32, 0, 1);
      __builtin_prefetch(deltas + g + 32, 0, 1);
      __builtin_prefetch(ts + g + 32, 0, 1);
      __builtin_prefetch(rgbs + 3LL * g + 96, 0, 1);
    }

    float sig = sigmas[g];
    float del = deltas[g];
    float t   = ts[g];
    const float* rg = rgbs + 3LL * g;
    float r0 = rg[0], g0 = rg[1], b0 = rg[2];

    process_chunk(lane, valid, g, sig, del, t, r0, g0, b0, thr,
                  carry, acc_op, acc_dep, acc_r, acc_g, acc_b, acc_cnt, out_w);
  }
#endif

  // Wave32 butterfly reduction of the six accumulators.
#pragma unroll
  for (int off = 16; off >= 1; off >>= 1) {
    acc_op  += __shfl_xor(acc_op,  off, 32);
    acc_dep += __shfl_xor(acc_dep, off, 32);
    acc_r   += __shfl_xor(acc_r,   off, 32);
    acc_g   += __shfl_xor(acc_g,   off, 32);
    acc_b   += __shfl_xor(acc_b,   off, 32);
    acc_cnt += __shfl_xor(acc_cnt, off, 32);
  }

  if (lane == 0) {
    out_op[ray_id]  = acc_op;
    out_dep[ray_id] = acc_dep;
    out_rgb[3 * ray_id + 0] = acc_r;
    out_rgb[3 * ray_id + 1] = acc_g;
    out_rgb[3 * ray_id + 2] = acc_b;
    // Counts are integers <= 2^24, exactly representable / exactly summed in f32.
    atomicAdd(out, (float)acc_cnt);
  }
}

extern "C" void kernel_launch(void* const* d_in, const int* in_sizes, int n_in,
                              void* d_out, int out_size, void* d_ws, size_t ws_size,
                              hipStream_t stream) {
  const float*     sigmas = (const float*)d_in[0];
  const float*     rgbs   = (const float*)d_in[1];
  const float*     deltas = (const float*)d_in[2];
  const float*     ts     = (const float*)d_in[3];
  const long long* rays_a = (const long long*)d_in[4];
  const float*     thr    = (const float*)d_in[5];
  float* out = (float*)d_out;

  int n_rays = in_sizes[4] / 3;

  zero_head_kernel<<<1, 1, 0, stream>>>(out);

  int blocks = (n_rays + WAVES_PER_BLOCK - 1) / WAVES_PER_BLOCK;
  volrend_fwd_kernel<<<blocks, BLOCK_THREADS, 0, stream>>>(
      sigmas, rgbs, deltas, ts, rays_a, thr, out, n_rays);
}